// TinyDIP_80178449481945
// MI455X (gfx1250) — compile-verified
//
#include <hip/hip_runtime.h>

// Histogram equalization (torchvision _scale_channel semantics) for
// x: (32, 3, 512, 512) fp32 in [0,1].  96 channels x 262144 pixels.
//
// Pipeline:
//   K0: zero per-channel histograms (96*256 ints in d_ws)
//   K1: per-wave privatized LDS histograms -> global atomics
//   K2: one wave per channel; 256-bin inclusive scan via 4x
//       V_WMMA_F32_16X16X4_F32 (P = H x UpperTriOnes, exact in fp32),
//       then build the final float LUT (identity when step==0)
//   K3: branch-free apply: out = lut[clamp(floor(x*255))]

typedef __attribute__((ext_vector_type(2))) float v2f;
typedef __attribute__((ext_vector_type(8))) float v8f;

__device__ __forceinline__ int quant255(float v) {
    int q = (int)floorf(v * 255.0f);
    q = q < 0 ? 0 : q;
    return q > 255 ? 255 : q;
}

// ---------------- K0: zero histograms ----------------
__global__ void eq_zero_hist(int* __restrict__ hist) {
    hist[blockIdx.x * 256 + threadIdx.x] = 0;
}

// ---------------- K1: histogram ----------------
// grid = nch * BLOCKS_PER_CH_HIST, block = 256 (8 wave32 waves)
#define BLOCKS_PER_CH_HIST 32
__global__ void eq_hist(const float* __restrict__ x, int* __restrict__ hist,
                        int vecPerChan) {
    __shared__ int lh[8][256];           // one sub-histogram per wave
    const int tid  = threadIdx.x;
    const int wave = tid >> 5;
    for (int i = tid; i < 8 * 256; i += blockDim.x) ((int*)lh)[i] = 0;
    __syncthreads();

    const int ch  = blockIdx.x / BLOCKS_PER_CH_HIST;
    const int blk = blockIdx.x % BLOCKS_PER_CH_HIST;
    const int vecPerBlock = vecPerChan / BLOCKS_PER_CH_HIST;   // 2048
    const float4* __restrict__ src =
        (const float4*)(x + (size_t)ch * vecPerChan * 4) + (size_t)blk * vecPerBlock;

    for (int i = tid; i < vecPerBlock; i += blockDim.x) {
        float4 p = src[i];
        atomicAdd(&lh[wave][quant255(p.x)], 1);
        atomicAdd(&lh[wave][quant255(p.y)], 1);
        atomicAdd(&lh[wave][quant255(p.z)], 1);
        atomicAdd(&lh[wave][quant255(p.w)], 1);
    }
    __syncthreads();

    for (int b = tid; b < 256; b += blockDim.x) {
        int s = 0;
        #pragma unroll
        for (int w = 0; w < 8; ++w) s += lh[w][b];
        if (s) atomicAdd(&hist[ch * 256 + b], s);
    }
}

// ---------------- K2: LUT build (WMMA prefix sum) ----------------
// grid = nch, block = 32 (exactly one wave32; EXEC all-ones through WMMA)
__global__ void eq_lut(const int* __restrict__ hist, float* __restrict__ lutf,
                       int npix) {
    __shared__ int   hsh[256];
    __shared__ float Pm[16][17];   // +1 pad: bank-conflict-free column reads
    __shared__ int   offs[16];
    __shared__ int   stepSh;

    const int ch   = blockIdx.x;
    const int lane = threadIdx.x;          // 0..31
    for (int i = lane; i < 256; i += 32) hsh[i] = hist[ch * 256 + i];
    __syncthreads();

    // H[r][c] = hist[r*16+c].  P = H(16x16) x U(16x16), U[k][j] = (k<=j).
    // Done as 4 accumulating 16x16x4 fp32 WMMAs (exact: counts <= 2^18).
    // f32 A-operand 16x4: VGPR j, lanes[0:15]->K=2*0+j ; lanes[16:31]->K=2+j
    // f32 B-operand 4x16: VGPR j, lanes[0:15]->row j   ; lanes[16:31]->row j+2
    const int half = lane >> 4;            // 0 or 1
    const int m    = lane & 15;
    v8f acc = {};
    #pragma unroll
    for (int t = 0; t < 4; ++t) {
        const int k0 = 4 * t + 2 * half;   // K index for VGPR0 of this lane
        v2f a, b;
        a.x = (float)hsh[m * 16 + k0];
        a.y = (float)hsh[m * 16 + k0 + 1];
        b.x = (k0     <= m) ? 1.0f : 0.0f; // U[k0  ][m]
        b.y = (k0 + 1 <= m) ? 1.0f : 0.0f; // U[k0+1][m]
        acc = __builtin_amdgcn_wmma_f32_16x16x4_f32(
            /*neg_a=*/false, a, /*neg_b=*/false, b,
            /*c_mod=*/(short)0, acc, /*reuse_a=*/false, /*reuse_b=*/false);
    }
    // D layout: VGPR v holds row (v + 8*half), column m
    #pragma unroll
    for (int v = 0; v < 8; ++v) Pm[v + 8 * half][m] = acc[v];
    __syncthreads();

    // step = (N - hist[last_nonzero]) // 255
    if (lane == 0) {
        int last = 0;
        for (int i = 0; i < 256; ++i)
            if (hsh[i] > 0) last = i;
        stepSh = (npix - hsh[last]) / 255;
    }
    // block offsets: offs[r] = sum_{s<r} rowTotal[s], rowTotal[s] = Pm[s][15]
    if (lane < 16) {
        int o = 0;
        for (int s = 0; s < lane; ++s) o += (int)Pm[s][15];
        offs[lane] = o;
    }
    __syncthreads();

    const int step  = stepSh;
    const int sstep = step > 0 ? step : 1;
    for (int i = lane; i < 256; i += 32) {
        float outv;
        if (step == 0) {
            outv = (float)i * (1.0f / 255.0f);     // identity mapping
        } else if (i == 0) {
            outv = 0.0f;                            // lut shifted: front-pad 0
        } else {
            const int j = i - 1;                    // inclusive cumsum at i-1
            const int cum = (int)Pm[j >> 4][j & 15] + offs[j >> 4];
            int l = (cum + (step >> 1)) / sstep;
            l = l < 0 ? 0 : (l > 255 ? 255 : l);
            outv = (float)l * (1.0f / 255.0f);
        }
        lutf[ch * 256 + i] = outv;
    }
}

// ---------------- K3: apply ----------------
// grid = nch * BLOCKS_PER_CH_APPLY, block = 256
#define BLOCKS_PER_CH_APPLY 64
__global__ void eq_apply(const float* __restrict__ x,
                         const float* __restrict__ lutf,
                         float* __restrict__ out, int vecPerChan) {
    __shared__ float l[256];
    const int ch  = blockIdx.x / BLOCKS_PER_CH_APPLY;
    const int blk = blockIdx.x % BLOCKS_PER_CH_APPLY;
    if (threadIdx.x < 256) l[threadIdx.x] = lutf[ch * 256 + threadIdx.x];
    __syncthreads();

    const int vecPerBlock = vecPerChan / BLOCKS_PER_CH_APPLY;  // 1024
    const size_t base = (size_t)ch * vecPerChan + (size_t)blk * vecPerBlock;
    const float4* __restrict__ src = (const float4*)x + base;
    float4* __restrict__ dst = (float4*)out + base;

    for (int i = threadIdx.x; i < vecPerBlock; i += blockDim.x) {
        float4 p = src[i];
        float4 q;
        q.x = l[quant255(p.x)];
        q.y = l[quant255(p.y)];
        q.z = l[quant255(p.z)];
        q.w = l[quant255(p.w)];
        dst[i] = q;
    }
}

extern "C" void kernel_launch(void* const* d_in, const int* in_sizes, int n_in,
                              void* d_out, int out_size, void* d_ws, size_t ws_size,
                              hipStream_t stream) {
    const float* x = (const float*)d_in[0];
    float* out = (float*)d_out;

    const int HW  = 512 * 512;                 // pixels per channel
    const int nch = in_sizes[0] / HW;          // 96 = 32*3
    const int vecPerChan = HW / 4;             // 65536 float4 per channel

    int*   hist = (int*)d_ws;                              // nch*256 ints
    float* lutf = (float*)((char*)d_ws + (size_t)nch * 256 * sizeof(int));

    eq_zero_hist<<<nch, 256, 0, stream>>>(hist);
    eq_hist<<<nch * BLOCKS_PER_CH_HIST, 256, 0, stream>>>(x, hist, vecPerChan);
    eq_lut<<<nch, 32, 0, stream>>>(hist, lutf, HW);
    eq_apply<<<nch * BLOCKS_PER_CH_APPLY, 256, 0, stream>>>(x, lutf, out, vecPerChan);
}